// NeuralGraphCollaborativeFiltering_34325378629786
// MI455X (gfx1250) — compile-verified
//
#include <hip/hip_runtime.h>
#include <hip/hip_bf16.h>

#define N_USERS 50000
#define N_ITEMS 150000
#define N_NODES (N_USERS + N_ITEMS)
#define N_EDGES 1250000
#define D 64
#define N_LAYERS 3
#define OUT_STRIDE ((N_LAYERS + 1) * D) // 256 floats per node in d_out

typedef __attribute__((ext_vector_type(2))) float v2f;
typedef __attribute__((ext_vector_type(4))) float v4f;
typedef __attribute__((ext_vector_type(8))) float v8f;

// ---------------------------------------------------------------------------
// e0 = concat(user_emb, item_emb) written into output stripe 0 (cols 0..63)
// ---------------------------------------------------------------------------
__global__ void k_init(const float* __restrict__ user_emb,
                       const float* __restrict__ item_emb,
                       float* __restrict__ out) {
    long tid = (long)blockIdx.x * blockDim.x + threadIdx.x; // one float4 each
    if (tid >= (long)N_NODES * (D / 4)) return;
    int node = (int)(tid >> 4);
    int c4   = (int)(tid & 15);
    const float* src = (node < N_USERS) ? (user_emb + (size_t)node * D)
                                        : (item_emb + (size_t)(node - N_USERS) * D);
    v4f v = *(const v4f*)(src + c4 * 4);
    *(v4f*)(out + (size_t)node * OUT_STRIDE + c4 * 4) = v;
}

// ---------------------------------------------------------------------------
// term1 = 0
// ---------------------------------------------------------------------------
__global__ void k_zero(float* __restrict__ buf, long n4) {
    long tid = (long)blockIdx.x * blockDim.x + threadIdx.x;
    if (tid >= n4) return;
    v4f z = {0.f, 0.f, 0.f, 0.f};
    *(v4f*)(buf + tid * 4) = z;
}

// ---------------------------------------------------------------------------
// SpMM scatter: term1[dst] += w * e[src].  16 lanes per edge, float4 per lane.
// e rows live inside d_out (stride OUT_STRIDE); term1 is dense [N, 64].
// Working set (e 51 MB + term1 51 MB) fits in the 192 MB L2, so the 80M
// fp32 atomics per layer resolve at L2 atomic throughput.
// ---------------------------------------------------------------------------
__global__ void k_spmm(const int* __restrict__ edge_src,
                       const int* __restrict__ edge_dst,
                       const float* __restrict__ edge_w,
                       const float* __restrict__ e,      // base of current stripe
                       float* __restrict__ term1) {
    long tid  = (long)blockIdx.x * blockDim.x + threadIdx.x;
    long edge = tid >> 4;
    if (edge >= N_EDGES) return;
    int c = (int)(tid & 15) * 4;
    int s = edge_src[edge];
    int d = edge_dst[edge];
    float w = edge_w[edge];
    v4f v = *(const v4f*)(e + (size_t)s * OUT_STRIDE + c);
    float* p = term1 + (size_t)d * D + c;
    atomicAdd(p + 0, w * v.x);
    atomicAdd(p + 1, w * v.y);
    atomicAdd(p + 2, w * v.z);
    atomicAdd(p + 3, w * v.w);
}

// ---------------------------------------------------------------------------
// Fused: inter = term1@W1 + (term1*e)@W2 + e@W3 + (b1+b2+b3)
//        e_next = l2norm(leaky_relu(inter))   -> written to next output stripe
// One wave per 16-node tile, V_WMMA_F32_16X16X4_F32, K unrolled 16x, 4 N-tiles.
//
// Weights are staged in LDS in a K-PAIRED layout:
//   lds_w[w][p][col][2] = { W[2p][col], W[2p+1][col] },  p = K-pair index.
// A B-operand for K-step k, half-wave h is then the contiguous 8-byte word at
// pair p = 2k+h, so each WMMA's B comes from ONE ds_load_b64 directly into an
// even-aligned VGPR pair (no repacking moves).
// ---------------------------------------------------------------------------
__global__ void __launch_bounds__(256)
k_gemm(const float* __restrict__ term1,
       const float* __restrict__ eCur,   // stride OUT_STRIDE
       const float* __restrict__ Wl,     // [3][64][64] for this layer
       const float* __restrict__ bl,     // [3][64] for this layer
       float* __restrict__ eNext) {      // stride OUT_STRIDE
    __shared__ float lds_w[3 * D * D];   // 48 KB, paired layout (see above)
    __shared__ float lds_b[D];           // summed bias

    // cooperative weight load: read float4 row-major, scatter into pair layout
    for (int j4 = threadIdx.x; j4 < 3 * D * D / 4; j4 += 256) {
        v4f v = *(const v4f*)(Wl + (size_t)j4 * 4);
        int w   = j4 >> 10;              // 1024 float4 per 64x64 matrix
        int rem = j4 & 1023;
        int r   = rem >> 4;              // source row
        int c0  = (rem & 15) * 4;        // first of 4 source cols
        float* base = lds_w + w * (D * D) + (r >> 1) * (2 * D) + (r & 1);
        base[(c0 + 0) * 2] = v.x;
        base[(c0 + 1) * 2] = v.y;
        base[(c0 + 2) * 2] = v.z;
        base[(c0 + 3) * 2] = v.w;
    }
    if (threadIdx.x < D)
        lds_b[threadIdx.x] = bl[threadIdx.x] + bl[D + threadIdx.x] + bl[2 * D + threadIdx.x];
    __syncthreads();

    int wid  = threadIdx.x >> 5;
    int lane = threadIdx.x & 31;
    int row_base = (blockIdx.x * 8 + wid) * 16;
    if (row_base >= N_NODES) return;     // whole wave drops together (EXEC all-1 for WMMA)

    int ln = lane & 15;                  // A row within tile / B,C column within tile
    int h  = lane >> 4;                  // half-wave selector

    // --- preload A operands for all 16 K-steps: lane holds (row ln, K=4k+2h..+1)
    const float* t1row = term1 + (size_t)(row_base + ln) * D;
    const float* erow  = eCur  + (size_t)(row_base + ln) * OUT_STRIDE;
    v2f a1[16], a2[16], ae[16];
#pragma unroll
    for (int k = 0; k < 16; ++k) {
        v2f x = *(const v2f*)(t1row + 4 * k + 2 * h);
        v2f y = *(const v2f*)(erow  + 4 * k + 2 * h);
        a1[k] = x;          // term1
        ae[k] = y;          // e
        a2[k] = x * y;      // term2 = term1 * e
    }

    // --- 4 N-tiles x 16 K-steps x 3 GEMMs = 192 WMMAs
    v8f acc[4];
#pragma unroll
    for (int t = 0; t < 4; ++t) {
        v8f c = {0.f, 0.f, 0.f, 0.f, 0.f, 0.f, 0.f, 0.f};
        int colx2 = (t * 16 + ln) * 2;
#pragma unroll
        for (int k = 0; k < 16; ++k) {
            int p = 2 * k + h;           // K-pair index for this half-wave
            const float* pb = lds_w + p * (2 * D) + colx2;
            v2f b1 = *(const v2f*)(pb + 0 * (D * D));
            c = __builtin_amdgcn_wmma_f32_16x16x4_f32(false, a1[k], false, b1,
                                                      (short)0, c, false, false);
            v2f b2 = *(const v2f*)(pb + 1 * (D * D));
            c = __builtin_amdgcn_wmma_f32_16x16x4_f32(false, a2[k], false, b2,
                                                      (short)0, c, false, false);
            v2f b3 = *(const v2f*)(pb + 2 * (D * D));
            c = __builtin_amdgcn_wmma_f32_16x16x4_f32(false, ae[k], false, b3,
                                                      (short)0, c, false, false);
        }
        acc[t] = c;
    }

    // --- bias + leaky_relu(0.01). C/D layout: VGPR v, half h -> row v+8h, col = ln
#pragma unroll
    for (int t = 0; t < 4; ++t) {
        float bias = lds_b[t * 16 + ln];
#pragma unroll
        for (int v = 0; v < 8; ++v) {
            float x = acc[t][v] + bias;
            acc[t][v] = (x > 0.f) ? x : 0.01f * x;
        }
    }

    // --- row-wise L2 normalize: reduce squares over the 16 lanes of each half
#pragma unroll
    for (int v = 0; v < 8; ++v) {
        float s = acc[0][v] * acc[0][v] + acc[1][v] * acc[1][v] +
                  acc[2][v] * acc[2][v] + acc[3][v] * acc[3][v];
        s += __shfl_xor(s, 1, 32);
        s += __shfl_xor(s, 2, 32);
        s += __shfl_xor(s, 4, 32);
        s += __shfl_xor(s, 8, 32);
        float inv = 1.0f / fmaxf(sqrtf(s), 1e-12f);
        float* orow = eNext + (size_t)(row_base + v + 8 * h) * OUT_STRIDE;
#pragma unroll
        for (int t = 0; t < 4; ++t)
            orow[t * 16 + ln] = acc[t][v] * inv;   // coalesced over the 16 lanes
    }
}

// ---------------------------------------------------------------------------
extern "C" void kernel_launch(void* const* d_in, const int* in_sizes, int n_in,
                              void* d_out, int out_size, void* d_ws, size_t ws_size,
                              hipStream_t stream) {
    (void)in_sizes; (void)n_in; (void)out_size; (void)ws_size;
    const int*   edge_src = (const int*)d_in[0];
    const int*   edge_dst = (const int*)d_in[1];
    const float* edge_w   = (const float*)d_in[2];
    const float* user_emb = (const float*)d_in[3];
    const float* item_emb = (const float*)d_in[4];
    const float* Ws       = (const float*)d_in[5]; // [3][3][64][64]
    const float* bs       = (const float*)d_in[6]; // [3][3][64]
    float* out   = (float*)d_out;                  // [N_NODES][256]
    float* term1 = (float*)d_ws;                   // [N_NODES][64] scratch

    {   // e0 stripe
        long n = (long)N_NODES * (D / 4);
        k_init<<<(int)((n + 255) / 256), 256, 0, stream>>>(user_emb, item_emb, out);
    }
    for (int layer = 0; layer < N_LAYERS; ++layer) {
        const float* eCur  = out + (size_t)layer * D;       // stripe `layer`
        float*       eNext = out + (size_t)(layer + 1) * D; // stripe `layer+1`

        long n4 = (long)N_NODES * D / 4;
        k_zero<<<(int)((n4 + 255) / 256), 256, 0, stream>>>(term1, n4);

        long spmm_threads = (long)N_EDGES * 16;
        k_spmm<<<(int)((spmm_threads + 255) / 256), 256, 0, stream>>>(
            edge_src, edge_dst, edge_w, eCur, term1);

        int tiles  = (N_NODES + 15) / 16;
        int blocks = (tiles + 7) / 8;          // 8 waves (tiles) per 256-thread block
        k_gemm<<<blocks, 256, 0, stream>>>(
            term1, eCur,
            Ws + (size_t)layer * 3 * D * D,
            bs + (size_t)layer * 3 * D,
            eNext);
    }
}